// NoisyTopItemsPerExpertRouter_90048284328017
// MI455X (gfx1250) — compile-verified
//
#include <hip/hip_runtime.h>

// Problem constants from the reference
#define G_ 8
#define S_ 2048
#define D_ 512
#define E_ 64
#define C_ 32

typedef __attribute__((ext_vector_type(2))) float v2f;
typedef __attribute__((ext_vector_type(8))) float v8f;

// ---------------------------------------------------------------------------
// Kernel 0: zero the (mostly sparse) 268MB output with wide b128 stores.
// ---------------------------------------------------------------------------
__global__ void zero_out(float* __restrict__ out, long n) {
    long i = (long)blockIdx.x * blockDim.x + threadIdx.x;
    long stride = (long)gridDim.x * blockDim.x;
    for (long j = i * 4; j < n; j += stride * 4) {
        *(float4*)(out + j) = make_float4(0.f, 0.f, 0.f, 0.f);
    }
}

// ---------------------------------------------------------------------------
// Kernel 1: gating GEMM  logits[g,s,e] = sum_d X[g,s,d] * W[d,e]
// One wave32 computes a 16(M) x 64(E) tile via V_WMMA_F32_16X16X4_F32.
//
// fp32 WMMA fragment layouts (ISA 7.12.2):
//   A (16x4, 2 VGPRs): lane<16 -> (row=lane,   K=k0,k0+1)
//                      lane>=16 -> (row=lane-16, K=k0+2,k0+3)
//   B (4x16, 2 VGPRs): mirrors A (lane<16 holds K=k0/k0+1 for col=lane, etc.)
//   C/D (16x16, 8 VGPRs): reg r -> (M=r | M=r+8), N=lane&15
//
// W is staged through LDS in 64-K chunks (16KB), PRE-SWIZZLED so that each
// lane's B fragment for (k-group kg, n-tile n) is one aligned ds_load_b64 at
//   lwsw[(kg*4 + n)*64 + lane*2]
// i.e. the 4 n-tiles of a k-group are consecutive 256B immediate offsets.
// All 16 A fragments of a chunk are issued up front to keep LOADcnt deep.
// ---------------------------------------------------------------------------
__global__ __launch_bounds__(256) void gemm_logits(const float* __restrict__ X,
                                                   const float* __restrict__ W,
                                                   float* __restrict__ logits) {
    __shared__ float lwsw[64 * E_];   // 16KB swizzled W chunk

    const int lane  = threadIdx.x & 31;
    const int wid   = threadIdx.x >> 5;            // 8 waves / block
    const int l15   = lane & 15;
    const int halfk = lane >> 4;                   // 0: K0/K1 half, 1: K2/K3 half
    const long rowBase = ((long)blockIdx.x * 8 + wid) * 16;   // 16 token rows

    v8f acc[4];
#pragma unroll
    for (int n = 0; n < 4; ++n) acc[n] = (v8f){0, 0, 0, 0, 0, 0, 0, 0};

    const float* xrow = X + (rowBase + l15) * D_;

    for (int kc = 0; kc < D_; kc += 64) {
        __syncthreads();   // previous chunk fully consumed
        // Cooperative staged+swizzled copy of W[kc..kc+63][0..63] (4096 floats).
        // Read coalesced float4 along E; scatter into B-fragment-native layout.
        for (int t4 = threadIdx.x; t4 < (64 * E_) / 4; t4 += 256) {
            float4 w4 = ((const float4*)(W + (long)kc * E_))[t4];
            if (kc + 64 < D_)   // prefetch next chunk (global_prefetch_b8)
                __builtin_prefetch(W + (long)(kc + 64) * E_ + t4 * 4, 0, 0);
            const int f  = t4 * 4;
            const int k  = f >> 6;     // 0..63 within chunk
            const int e0 = f & 63;
            const int kg = k >> 2, kr = k & 3;
            const int dl = (kr >> 1) * 16;       // dest lane base (halfk*16)
            const int comp = kr & 1;
            const float wv[4] = {w4.x, w4.y, w4.z, w4.w};
#pragma unroll
            for (int i = 0; i < 4; ++i) {
                const int e = e0 + i;
                lwsw[((kg * 4 + (e >> 4)) * 64) + (dl + (e & 15)) * 2 + comp] = wv[i];
            }
        }
        __syncthreads();

        // Issue ALL 16 A fragments of this chunk first (deep LOADcnt pipe).
        v2f afrag[16];
#pragma unroll
        for (int j = 0; j < 16; ++j)
            afrag[j] = *(const v2f*)(xrow + kc + j * 4 + halfk * 2);

        // 16 k-groups x 4 n-tiles of WMMA; B frags are single ds_load_b64s.
#pragma unroll
        for (int kg = 0; kg < 16; ++kg) {
            const float* bbase = lwsw + kg * 256 + lane * 2;
            v2f b0 = *(const v2f*)(bbase + 0);
            v2f b1 = *(const v2f*)(bbase + 64);
            v2f b2 = *(const v2f*)(bbase + 128);
            v2f b3 = *(const v2f*)(bbase + 192);
            acc[0] = __builtin_amdgcn_wmma_f32_16x16x4_f32(false, afrag[kg], false, b0,
                                                           (short)0, acc[0], false, false);
            acc[1] = __builtin_amdgcn_wmma_f32_16x16x4_f32(false, afrag[kg], false, b1,
                                                           (short)0, acc[1], false, false);
            acc[2] = __builtin_amdgcn_wmma_f32_16x16x4_f32(false, afrag[kg], false, b2,
                                                           (short)0, acc[2], false, false);
            acc[3] = __builtin_amdgcn_wmma_f32_16x16x4_f32(false, afrag[kg], false, b3,
                                                           (short)0, acc[3], false, false);
        }
    }

    // Write back the 16x64 logits tile per the C/D layout.
    const long outRow = rowBase + (halfk ? 8 : 0);
#pragma unroll
    for (int r = 0; r < 8; ++r) {
#pragma unroll
        for (int n = 0; n < 4; ++n) {
            logits[(outRow + r) * E_ + n * 16 + l15] = acc[n][r];
        }
    }
}

// ---------------------------------------------------------------------------
// Kernel 2: noise + softmax, in-place on the logits buffer (one thread/token).
// JAX threefry noise stream is not reproducible bit-exactly; use a counter-
// based hash + Box-Muller with the same std (NOISE_STD/NUM_EXPERTS = 1/64).
// ---------------------------------------------------------------------------
__device__ __forceinline__ unsigned pcg_hash(unsigned v) {
    v = v * 747796405u + 2891336453u;
    v = ((v >> ((v >> 28u) + 4u)) ^ v) * 277803737u;
    return (v >> 22u) ^ v;
}

__global__ __launch_bounds__(256) void noisy_softmax(float* __restrict__ gates) {
    long row = (long)blockIdx.x * blockDim.x + threadIdx.x;
    if (row >= (long)G_ * S_) return;

    float v[E_];
    float mx = -1e30f;
#pragma unroll 8
    for (int e = 0; e < E_; ++e) {
        float x = gates[row * E_ + e];
        unsigned idx = (unsigned)(row * E_ + e);
        unsigned h1 = pcg_hash(idx * 2u + 1u);
        unsigned h2 = pcg_hash((idx * 2u + 2u) ^ 0x9E3779B9u);
        float u1 = ((float)h1 + 1.0f) * (1.0f / 4294967296.0f);  // (0,1]
        float u2 = ((float)h2) * (1.0f / 4294967296.0f);         // [0,1)
        float nrm = sqrtf(-2.0f * __logf(u1)) * __cosf(6.28318530718f * u2);
        x += (1.0f / 64.0f) * nrm;
        v[e] = x;
        mx = fmaxf(mx, x);
    }
    float s = 0.0f;
#pragma unroll 8
    for (int e = 0; e < E_; ++e) { v[e] = __expf(v[e] - mx); s += v[e]; }
    float inv = 1.0f / s;
#pragma unroll 8
    for (int e = 0; e < E_; ++e) gates[row * E_ + e] = v[e] * inv;
}

// ---------------------------------------------------------------------------
// Kernel 3: experts-choice top-C per (group, expert). One block per pair.
// Gates column cached in LDS (8KB); 32 rounds of tie-stable argmax
// (ties -> lowest token index, matching lax.top_k); scatter outputs.
//   combine [g][s][e][c] = gate   dispatch [g][e][c][s] = 1
// ---------------------------------------------------------------------------
__global__ __launch_bounds__(256) void topk_dispatch(const float* __restrict__ gates,
                                                     float* __restrict__ combine,
                                                     float* __restrict__ dispatch) {
    __shared__ float vals[S_];
    __shared__ float rv[256];
    __shared__ int   ri[256];

    const int g = blockIdx.x / E_;
    const int e = blockIdx.x % E_;
    const int t = threadIdx.x;

    for (int s = t; s < S_; s += 256)
        vals[s] = gates[(long)(g * S_ + s) * E_ + e];
    __syncthreads();

    for (int c = 0; c < C_; ++c) {
        // local scan (increasing s => keeps lowest index on ties)
        float bv = -1e30f; int bi = S_;
        for (int s = t; s < S_; s += 256) {
            float x = vals[s];
            if (x > bv) { bv = x; bi = s; }
        }
        rv[t] = bv; ri[t] = bi;
        __syncthreads();
        // tree reduce 256 -> 1, ties prefer lower token index
        for (int off = 128; off > 0; off >>= 1) {
            if (t < off) {
                float ov = rv[t + off]; int oi = ri[t + off];
                if (ov > rv[t] || (ov == rv[t] && oi < ri[t])) { rv[t] = ov; ri[t] = oi; }
            }
            __syncthreads();
        }
        if (t == 0) {
            int   s   = ri[0];
            float val = rv[0];
            vals[s] = -1e38f;   // remove from further rounds
            combine[((long)(g * S_ + s) * E_ + e) * C_ + c] = val;
            dispatch[((long)(g * E_ + e) * C_ + c) * S_ + s] = 1.0f;
        }
        __syncthreads();
    }
}

// ---------------------------------------------------------------------------
extern "C" void kernel_launch(void* const* d_in, const int* in_sizes, int n_in,
                              void* d_out, int out_size, void* d_ws, size_t ws_size,
                              hipStream_t stream) {
    const float* X = (const float*)d_in[0];   // (G, S, D) fp32
    const float* W = (const float*)d_in[1];   // (D, E)    fp32
    float* out = (float*)d_out;
    float* ws  = (float*)d_ws;                // logits/gates: G*S*E fp32 = 4MB

    const long combineN = (long)G_ * S_ * E_ * C_;   // 33,554,432
    const long totalN   = combineN * 2;              // + dispatch

    // 1) zero the sparse outputs (write-bound floor of the whole op)
    zero_out<<<2048, 256, 0, stream>>>(out, totalN);
    // 2) gating GEMM via fp32 WMMA (128 blocks x 8 waves x 16 rows = 16384)
    gemm_logits<<<(G_ * S_) / 128, 256, 0, stream>>>(X, W, ws);
    // 3) noise + softmax in-place (one thread per token)
    noisy_softmax<<<((G_ * S_) + 255) / 256, 256, 0, stream>>>(ws);
    // 4) experts-choice top-32 + scatter (512 blocks, one per (g,e))
    topk_dispatch<<<G_ * E_, 256, 0, stream>>>(ws, out, out + combineN);
}